// ToeplitzLinearAttention_11398843203810
// MI455X (gfx1250) — compile-verified
//
#include <hip/hip_runtime.h>

#define BATCH 4
#define SEQ   1024
#define NDIM  384
#define HEADS 12
#define HDIM  32
#define EPSF  1e-6f

typedef float v2f __attribute__((ext_vector_type(2)));
typedef float v8f __attribute__((ext_vector_type(8)));

// D = A(16x4,f32) * B(4x16,f32) + C(16x16,f32); full fp32 matrix op on CDNA5.
__device__ __forceinline__ v8f wmma_f32(v2f a, v2f b, v8f c) {
  return __builtin_amdgcn_wmma_f32_16x16x4_f32(false, a, false, b, (short)0, c,
                                               false, false);
}

// Async-copy one contiguous 2048B tile (16 rows x 32 f32) global -> LDS.
// 32 lanes x 16B = 512B per GLOBAL_LOAD_ASYNC_TO_LDS_B128; 4 issues per tile.
// Tracked by ASYNCcnt (completion is in-order for async loads).
__device__ __forceinline__ void async_tile_2048(const float* gsrc,
                                                float* lds_dst, int lane) {
  unsigned lbase = (unsigned)(size_t)lds_dst + (unsigned)lane * 16u;
  const char* g = (const char*)gsrc + lane * 16;
  #pragma unroll
  for (int t = 0; t < 4; ++t) {
    unsigned loff = lbase + (unsigned)t * 512u;
    const char* ga = g + t * 512;
    asm volatile("global_load_async_to_lds_b128 %0, %1, off"
                 :: "v"(loff), "v"(ga) : "memory");
  }
}

// ---------------------------------------------------------------------------
// Kernel 1: QKV = X @ W_qkv, fused ReLU+eps on Q/K, scatter to [BH][N][32]
// grid (72, 256), block 32 (one wave per 16x16 output tile)
// ---------------------------------------------------------------------------
__global__ void qkv_proj_kernel(const float* __restrict__ X,
                                const float* __restrict__ W,
                                float* __restrict__ Qo,
                                float* __restrict__ Ko,
                                float* __restrict__ Vo) {
  const int lane = threadIdx.x;
  const int l    = lane & 15;
  const int half = lane >> 4;
  const int tn   = blockIdx.x * 16;   // 0..1151
  const int tm   = blockIdx.y * 16;   // 0..4095

  const int arow = (tm + l) * NDIM;
  const int kofs = 2 * half;

  v8f c = {};
  #pragma unroll 4
  for (int kk = 0; kk < NDIM; kk += 4) {
    v2f a = *(const v2f*)(X + arow + kk + kofs);      // global_load_b64
    v2f b;
    b.x = W[(kk + kofs + 0) * (3 * NDIM) + tn + l];
    b.y = W[(kk + kofs + 1) * (3 * NDIM) + tn + l];
    c = wmma_f32(a, b, c);
  }

  const int part = tn / NDIM;                 // wave-uniform: 0=Q 1=K 2=V
  const int rem  = tn + l - part * NDIM;
  const int h    = rem >> 5;
  const int dc   = rem & 31;
  float* dst = (part == 0) ? Qo : (part == 1 ? Ko : Vo);
  const bool act = (part < 2);

  #pragma unroll
  for (int r = 0; r < 8; ++r) {
    const int mrow = tm + r + 8 * half;       // C layout: M = r + 8*half
    const int bb   = mrow >> 10;
    const int n    = mrow & (SEQ - 1);
    float v = c[r];
    if (act) v = fmaxf(v, 0.0f) + EPSF;       // relu(q/k) + eps
    dst[(((size_t)bb * HEADS + h) * SEQ + n) * HDIM + dc] = v;
  }
}

// ---------------------------------------------------------------------------
// Kernel 2: Toeplitz-masked linear attention, flash-style, async-staged K/V.
//   S = (Q K^T) ∘ T,  num += S V,  den += S·1,  O = num/(den+eps)
// grid (32, 48), block 32: one wave handles 32 queries (2 M-subtiles) of one
// head; K/V tiles double-buffered in LDS via GLOBAL_LOAD_ASYNC_TO_LDS_B128.
// ---------------------------------------------------------------------------
__global__ void toeplitz_attn_kernel(const float* __restrict__ Q,
                                     const float* __restrict__ K,
                                     const float* __restrict__ V,
                                     const float* __restrict__ tw,
                                     float* __restrict__ O) {
  __shared__ float lds_k[2][16 * HDIM];   // 2 x 2KB K tiles
  __shared__ float lds_v[2][16 * HDIM];   // 2 x 2KB V tiles
  __shared__ float lds_s[32 * 18];        // S tiles, stride 18 (align + banks)
  __shared__ float lds_d[64];
  __shared__ float lds_df[32];

  const int lane = threadIdx.x;
  const int l    = lane & 15;
  const int half = lane >> 4;
  const int q0   = blockIdx.x * 32;
  const int bh   = blockIdx.y;
  const int h    = bh % HEADS;
  const int b    = bh / HEADS;

  const float* Qh = Q + (size_t)bh * SEQ * HDIM;
  const float* Kh = K + (size_t)bh * SEQ * HDIM;
  const float* Vh = V + (size_t)bh * SEQ * HDIM;

  // Q tiles as A-fragments (2 subtiles x 8 chunks), resident all kernel.
  v2f aq[2][8];
  #pragma unroll
  for (int ms = 0; ms < 2; ++ms)
    #pragma unroll
    for (int t = 0; t < 8; ++t)
      aq[ms][t] =
          *(const v2f*)(Qh + (q0 + ms * 16 + l) * HDIM + t * 4 + 2 * half);

  v8f num[2][2] = {{{}, {}}, {{}, {}}};
  float den[2] = {0.0f, 0.0f};

  // Prime buffer 0 with the first K/V tile (8 async ops outstanding).
  async_tile_2048(Kh, &lds_k[0][0], lane);
  async_tile_2048(Vh, &lds_v[0][0], lane);

  for (int j0 = 0; j0 < SEQ; j0 += 16) {
    const int cur = (j0 >> 4) & 1;

    if (j0 + 16 < SEQ) {
      // WAR: make sure prior ds reads of buf cur^1 have drained.
      asm volatile("s_wait_dscnt 0x0" ::: "memory");
      async_tile_2048(Kh + (j0 + 16) * HDIM, &lds_k[cur ^ 1][0], lane);
      async_tile_2048(Vh + (j0 + 16) * HDIM, &lds_v[cur ^ 1][0], lane);
      // 8 newly issued remain; wait for the current tile's 8 to complete.
      asm volatile("s_wait_asynccnt 0x8" ::: "memory");
    } else {
      asm volatile("s_wait_asynccnt 0x0" ::: "memory");
    }
    __syncthreads();

    const float* kb = &lds_k[cur][0];
    const float* vb = &lds_v[cur][0];
    const int jcol = j0 + l;

    // ---- S = Q * K^T (d=32 -> 8 WMMAs per subtile), mask, stash to LDS ----
    #pragma unroll
    for (int ms = 0; ms < 2; ++ms) {
      v8f s = {};
      #pragma unroll
      for (int t = 0; t < 8; ++t) {
        const int kc = t * 4 + 2 * half;
        v2f bk = *(const v2f*)(kb + l * HDIM + kc);   // ds_load_b64
        s = wmma_f32(aq[ms][t], bk, s);
      }
      #pragma unroll
      for (int r = 0; r < 8; ++r) {
        const int irow = q0 + ms * 16 + r + 8 * half;
        s[r] *= tw[irow - jcol + (SEQ - 1)];          // Toeplitz mask
      }
      #pragma unroll
      for (int r = 0; r < 8; ++r)
        lds_s[(ms * 16 + r + 8 * half) * 18 + l] = s[r];
    }
    __syncthreads();

    // ---- num += S' V (A-frags re-read from LDS); fold row sums (den) ----
    #pragma unroll
    for (int ms = 0; ms < 2; ++ms) {
      #pragma unroll
      for (int t = 0; t < 4; ++t) {
        const int kc = t * 4 + 2 * half;
        v2f as = *(const v2f*)(&lds_s[(ms * 16 + l) * 18 + kc]);
        den[ms] += as.x + as.y;
        v2f bv0, bv1;                       // B[k][n] = V[j0+k][n(+16)]
        bv0.x = vb[(kc + 0) * HDIM + l];
        bv0.y = vb[(kc + 1) * HDIM + l];
        bv1.x = vb[(kc + 0) * HDIM + 16 + l];
        bv1.y = vb[(kc + 1) * HDIM + 16 + l];
        num[ms][0] = wmma_f32(as, bv0, num[ms][0]);
        num[ms][1] = wmma_f32(as, bv1, num[ms][1]);
      }
    }
    __syncthreads();
  }

  // combine den halves: lanes (m) and (m+16) hold partials for row m
  lds_d[lane]      = den[0];
  lds_d[32 + lane] = den[1];
  __syncthreads();
  if (lane < 16) {
    lds_df[lane]      = lds_d[lane]      + lds_d[lane + 16];
    lds_df[16 + lane] = lds_d[32 + lane] + lds_d[48 + lane];
  }
  __syncthreads();

  #pragma unroll
  for (int ms = 0; ms < 2; ++ms) {
    #pragma unroll
    for (int r = 0; r < 8; ++r) {
      const int m   = ms * 16 + r + 8 * half;
      const float d = lds_df[m] + EPSF;
      const int row = (b << 10) + q0 + m;
      const size_t base = (size_t)row * NDIM + h * HDIM;
      O[base + l]      = num[ms][0][r] / d;
      O[base + 16 + l] = num[ms][1][r] / d;
    }
  }
}

// ---------------------------------------------------------------------------
// Kernel 3: Out = A @ w_out^T + b_out       (A: [4096,384], w_out: [384,384])
// grid (24, 256), block 32
// ---------------------------------------------------------------------------
__global__ void out_proj_kernel(const float* __restrict__ A,
                                const float* __restrict__ Wo,
                                const float* __restrict__ Bo,
                                float* __restrict__ Out) {
  const int lane = threadIdx.x;
  const int l    = lane & 15;
  const int half = lane >> 4;
  const int tn   = blockIdx.x * 16;
  const int tm   = blockIdx.y * 16;

  v8f c = {};
  #pragma unroll 4
  for (int kk = 0; kk < NDIM; kk += 4) {
    const int kc = kk + 2 * half;
    v2f a = *(const v2f*)(A + (tm + l) * NDIM + kc);    // global_load_b64
    v2f b = *(const v2f*)(Wo + (tn + l) * NDIM + kc);   // B[k][n] = w_out[n][k]
    c = wmma_f32(a, b, c);
  }

  const float bias = Bo[tn + l];
  #pragma unroll
  for (int r = 0; r < 8; ++r)
    Out[(size_t)(tm + r + 8 * half) * NDIM + tn + l] = c[r] + bias;
}

// ---------------------------------------------------------------------------
extern "C" void kernel_launch(void* const* d_in, const int* in_sizes, int n_in,
                              void* d_out, int out_size, void* d_ws,
                              size_t ws_size, hipStream_t stream) {
  const float* x    = (const float*)d_in[0];   // [4,1024,384]
  const float* wqkv = (const float*)d_in[1];   // [384,1152]
  const float* wout = (const float*)d_in[2];   // [384,384]
  const float* bout = (const float*)d_in[3];   // [384]
  const float* tw   = (const float*)d_in[4];   // [2047]
  float* out = (float*)d_out;                  // [4,1024,384]

  float* ws = (float*)d_ws;
  const size_t per = (size_t)BATCH * HEADS * SEQ * HDIM;  // 1,572,864 floats
  float* Q   = ws;
  float* K   = Q + per;
  float* V   = K + per;
  float* ATT = V + per;                                   // [4096,384]

  qkv_proj_kernel<<<dim3(3 * NDIM / 16, BATCH * SEQ / 16), 32, 0, stream>>>(
      x, wqkv, Q, K, V);
  toeplitz_attn_kernel<<<dim3(SEQ / 32, BATCH * HEADS), 32, 0, stream>>>(
      Q, K, V, tw, ATT);
  out_proj_kernel<<<dim3(NDIM / 16, BATCH * SEQ / 16), 32, 0, stream>>>(
      ATT, wout, bout, out);
}